// Skeletonize_38139309588631
// MI455X (gfx1250) — compile-verified
//
#include <hip/hip_runtime.h>
#include <hip/hip_bf16.h>

// Skeletonize: threshold -> binary closing (disk(2) == plus ⊕ plus) ->
// 24 x (2 Zhang-Suen sub-iterations), on a bit-packed mask resident in LDS.
// One workgroup (1024 threads / 32 waves) per 512x512 slice.
// Each thread owns a vertical strip: column c = tid&15, rows [ (tid>>4)*8, +8 ).
// Rolling 3-row register window carries (center, east, west) bit-planes; only the
// strip-boundary rows pay range-guard VALU, interior loads are unguarded.

#define ROWS 512
#define WPR 16                  // data words per row (512/32)
#define ST 18                   // padded row stride in words: half-waves hit disjoint bank sets
#define BUFW (1 + ROWS * ST)    // +1 leading pad so (r=0,c=0) west read hits a zero word
#define PIX_PER_SLICE (512 * 512)
#define NTHREADS 1024
#define STRIPH 8                // rows per thread strip
#define N_ITER 24

static __device__ __forceinline__ unsigned ballot32(int p) {
#if __has_builtin(__builtin_amdgcn_ballot_w32)
    return __builtin_amdgcn_ballot_w32(p);
#else
    return (unsigned)__ballot(p);   // wave32: low 32 bits valid
#endif
}

// Unguarded row-plane load: r is statically known to be in [0, 511].
// Columns 16/17 of every row and word 0 are permanent zero pads, so the
// c+1 / c-1 reads are unconditional.
static __device__ __forceinline__ void load_row_u(const unsigned* __restrict__ B, int r, int c,
                                                  unsigned& cc, unsigned& ee, unsigned& ww) {
    const unsigned* row = B + 1 + r * ST;
    const unsigned ctr = row[c];
    const unsigned rt  = row[c + 1];
    const unsigned lt  = row[c - 1];
    cc = ctr;
    ee = (ctr >> 1) | (rt << 31);   // east neighbor of pixel b -> bit b
    ww = (ctr << 1) | (lt >> 31);   // west neighbor
}

// Guarded row-plane load for strip-boundary rows (r may be -1 or 512):
// clamp keeps the LDS address legal, masks give the zero-padding semantics.
static __device__ __forceinline__ void load_row_g(const unsigned* __restrict__ B, int r, int c,
                                                  unsigned& cc, unsigned& ee, unsigned& ww) {
    const int rcl = r < 0 ? 0 : (r > (ROWS - 1) ? (ROWS - 1) : r);
    const unsigned* row = B + 1 + rcl * ST;
    unsigned ctr = row[c];
    unsigned rt  = row[c + 1];
    unsigned lt  = row[c - 1];
    const bool ok = ((unsigned)r < (unsigned)ROWS);
    ctr = ok ? ctr : 0u;
    rt  = ok ? rt  : 0u;
    lt  = ok ? lt  : 0u;
    cc = ctr;
    ee = (ctr >> 1) | (rt << 31);
    ww = (ctr << 1) | (lt >> 31);
}

// Zhang-Suen deletion logic on a 32-pixel word (bit-sliced boolean logic).
static __device__ __forceinline__ unsigned zs_logic(unsigned w, unsigned N, unsigned NE,
                                                    unsigned E, unsigned SE, unsigned S,
                                                    unsigned SW, unsigned Wn, unsigned NW,
                                                    int step) {
    // cB: 2 <= B <= 6  <=>  (>=2 neighbors ON) AND (>=2 neighbors OFF)
    unsigned g1, g2, o1, o2, q;
    g1 = N;  g2 = 0;
    g2 |= g1 & NE; g1 |= NE;
    g2 |= g1 & E;  g1 |= E;
    g2 |= g1 & SE; g1 |= SE;
    g2 |= g1 & S;  g1 |= S;
    g2 |= g1 & SW; g1 |= SW;
    g2 |= g1 & Wn; g1 |= Wn;
    g2 |= g1 & NW; g1 |= NW;
    o1 = ~N; o2 = 0;
    q = ~NE; o2 |= o1 & q; o1 |= q;
    q = ~E;  o2 |= o1 & q; o1 |= q;
    q = ~SE; o2 |= o1 & q; o1 |= q;
    q = ~S;  o2 |= o1 & q; o1 |= q;
    q = ~SW; o2 |= o1 & q; o1 |= q;
    q = ~Wn; o2 |= o1 & q; o1 |= q;
    q = ~NW; o2 |= o1 & q; o1 |= q;
    const unsigned cB = g2 & o2;

    // cA: exactly one 0->1 transition around the ring N,NE,E,SE,S,SW,W,NW,N
    unsigned a1 = 0, a2 = 0, t;
    t = ~N  & NE; a2 |= a1 & t; a1 |= t;
    t = ~NE & E;  a2 |= a1 & t; a1 |= t;
    t = ~E  & SE; a2 |= a1 & t; a1 |= t;
    t = ~SE & S;  a2 |= a1 & t; a1 |= t;
    t = ~S  & SW; a2 |= a1 & t; a1 |= t;
    t = ~SW & Wn; a2 |= a1 & t; a1 |= t;
    t = ~Wn & NW; a2 |= a1 & t; a1 |= t;
    t = ~NW & N;  a2 |= a1 & t; a1 |= t;
    const unsigned cA = a1 & ~a2;

    unsigned c3, c4;
    if (step == 0) { c3 = ~(N & E & S);  c4 = ~(E & S & Wn); }
    else           { c3 = ~(N & E & Wn); c4 = ~(N & S & Wn); }

    return w & ~(cB & cA & c3 & c4);
}

// One Zhang-Suen sub-iteration over a thread's strip, rolling row window.
static __device__ __forceinline__ void thin_pass(const unsigned* __restrict__ Sb,
                                                 unsigned* __restrict__ Db,
                                                 int c, int r0, int step) {
    unsigned cm, em, wm;   // row r-1 planes
    unsigned c0, e0, w0;   // row r   planes
    unsigned cp, ep, wp;   // row r+1 planes
    load_row_g(Sb, r0 - 1, c, cm, em, wm);   // may be row -1
    load_row_u(Sb, r0,     c, c0, e0, w0);
#pragma unroll
    for (int k = 0; k < STRIPH; ++k) {
        const int r = r0 + k;
        if (k < STRIPH - 1) load_row_u(Sb, r + 1, c, cp, ep, wp);   // rows r0+1..r0+7: in range
        else                load_row_g(Sb, r + 1, c, cp, ep, wp);   // row r0+8: may be 512
        // N=cm NE=em E=e0 SE=ep S=cp SW=wp W=w0 NW=wm
        Db[1 + r * ST + c] = zs_logic(c0, cm, em, e0, ep, cp, wp, w0, wm, step);
        cm = c0; em = e0; wm = w0;
        c0 = cp; e0 = ep; w0 = wp;
    }
}

// One plus-shaped (3x3 cross) dilation/erosion over a strip.
static __device__ __forceinline__ void morph_pass(const unsigned* __restrict__ Sb,
                                                  unsigned* __restrict__ Db,
                                                  int c, int r0, bool dil) {
    unsigned cm, em, wm, c0, e0, w0, cp, ep, wp;
    load_row_g(Sb, r0 - 1, c, cm, em, wm);
    load_row_u(Sb, r0,     c, c0, e0, w0);
#pragma unroll
    for (int k = 0; k < STRIPH; ++k) {
        const int r = r0 + k;
        if (k < STRIPH - 1) load_row_u(Sb, r + 1, c, cp, ep, wp);
        else                load_row_g(Sb, r + 1, c, cp, ep, wp);
        Db[1 + r * ST + c] = dil ? (c0 | cm | cp | e0 | w0)
                                 : (c0 & cm & cp & e0 & w0);
        cm = c0; em = e0; wm = w0;
        c0 = cp; e0 = ep; w0 = wp;
    }
}

__global__ __launch_bounds__(NTHREADS)
void Skeletonize_38139309588631_kernel(const float* __restrict__ in, float* __restrict__ out) {
    __shared__ unsigned bufA[BUFW];
    __shared__ unsigned bufB[BUFW];

    const int z = blockIdx.x;
    const float* src = in  + (size_t)z * PIX_PER_SLICE;
    float*       dst = out + (size_t)z * PIX_PER_SLICE;
    const int tid  = threadIdx.x;
    const int lane = tid & 31;
    const int wv   = tid >> 5;
    const int col  = tid & 15;
    const int r0   = (tid >> 4) * STRIPH;

    // ---- zero both buffers once: establishes permanent zero pads ----
    for (int i = tid; i < BUFW; i += NTHREADS) { bufA[i] = 0u; bufB[i] = 0u; }
    __syncthreads();

    // ---- threshold + bit-pack (wave32 ballot: one packed word per step) ----
    for (int it = 0; it < 256; ++it) {
        const int wi  = (wv << 8) + it;            // word id 0..8191 (r = wi>>4, c = wi&15)
        const int pix = (wi << 5) + lane;
        int pf = pix + 8192;                       // stream-prefetch 32KB ahead, branchless
        pf = pf < PIX_PER_SLICE ? pf : PIX_PER_SLICE - 1;
        __builtin_prefetch(src + pf, 0, 0);        // global_prefetch_b8
        const unsigned m = ballot32(src[pix] > 0.5f);
        if (lane == 0) bufA[1 + (wi >> 4) * ST + (wi & 15)] = m;
    }
    __syncthreads();

    // ---- closing: dilate(plus) x2 then erode(plus) x2 (disk(2) = plus ⊕ plus) ----
    morph_pass(bufA, bufB, col, r0, true);  __syncthreads();
    morph_pass(bufB, bufA, col, r0, true);  __syncthreads();
    morph_pass(bufA, bufB, col, r0, false); __syncthreads();
    morph_pass(bufB, bufA, col, r0, false); __syncthreads();
    // closed mask in bufA

    // ---- Zhang-Suen thinning: 24 x (sub0: A->B, sub1: B->A) ----
    for (int iter = 0; iter < N_ITER; ++iter) {
        thin_pass(bufA, bufB, col, r0, 0); __syncthreads();
        thin_pass(bufB, bufA, col, r0, 1); __syncthreads();
    }

    // ---- unpack bits -> float output (LDS broadcast read, coalesced stores) ----
    for (int it = 0; it < 256; ++it) {
        const int wi = (wv << 8) + it;
        const unsigned m = bufA[1 + (wi >> 4) * ST + (wi & 15)];  // broadcast across lanes
        dst[(wi << 5) + lane] = ((m >> lane) & 1u) ? 1.0f : 0.0f;
    }
}

extern "C" void kernel_launch(void* const* d_in, const int* in_sizes, int n_in,
                              void* d_out, int out_size, void* d_ws, size_t ws_size,
                              hipStream_t stream) {
    const float* in = (const float*)d_in[0];
    float* out = (float*)d_out;
    const int depth = in_sizes[0] / PIX_PER_SLICE;   // 64 slices
    Skeletonize_38139309588631_kernel<<<dim3(depth), dim3(NTHREADS), 0, stream>>>(in, out);
}